// TotalLossVariable_83425444757832
// MI455X (gfx1250) — compile-verified
//
#include <hip/hip_runtime.h>

// CDNA5 / gfx1250 implementation of the NURBS surface loss.
// fp32 WMMA (V_WMMA_F32_16X16X4_F32) for both matmul stages of the einsum.

typedef __attribute__((ext_vector_type(2))) float v2f;
typedef __attribute__((ext_vector_type(8))) float v8f;

#define P_DEG   3
#define MU_MAX  32
#define OUT_N   64
#define B_SZ    1024
#define SURF_EPS 1e-5f

// ---------------------------------------------------------------- init ws
__global__ void init_ws_kernel(float* ws) {
    if (threadIdx.x < 4) ws[threadIdx.x] = 0.0f;   // ws[3] doubles as int nan-flag (bits 0)
}

// ------------------------------------------------------- masked ctrl MSE
__global__ __launch_bounds__(256)
void ctrl_loss_kernel(const float* __restrict__ pred,
                      const float* __restrict__ gt,
                      const float* __restrict__ mask,
                      float* __restrict__ ws) {
    __shared__ float sA[256];
    __shared__ float sB[256];
    float accD = 0.0f, accM = 0.0f;
    const int n = B_SZ * MU_MAX * MU_MAX;
    for (int idx = blockIdx.x * blockDim.x + threadIdx.x; idx < n;
         idx += gridDim.x * blockDim.x) {
        float m = mask[idx];
        const float* p = pred + idx * 3;
        const float* g = gt   + idx * 3;
        float d0 = p[0] - g[0];
        float d1 = p[1] - g[1];
        float d2 = p[2] - g[2];
        accD += (d0 * d0 + d1 * d1 + d2 * d2) * m;
        accM += m;
    }
    sA[threadIdx.x] = accD;
    sB[threadIdx.x] = accM;
    __syncthreads();
    for (int s = 128; s > 0; s >>= 1) {
        if ((int)threadIdx.x < s) {
            sA[threadIdx.x] += sA[threadIdx.x + s];
            sB[threadIdx.x] += sB[threadIdx.x + s];
        }
        __syncthreads();
    }
    if (threadIdx.x == 0) {
        atomicAdd(&ws[0], sA[0]);
        atomicAdd(&ws[1], sB[0]);
    }
}

// ------------------------------------------- per-sample surface eval loss
// One 256-thread block (8 wave32) per batch sample.
__global__ __launch_bounds__(256)
void surf_loss_kernel(const float* __restrict__ ctrl,
                      const float* __restrict__ mask,
                      const float* __restrict__ xyz,
                      float* __restrict__ ws) {
    __shared__ float sBu[64][32];     // basis over u           (8 KB)
    __shared__ float sBv[64][32];     // basis over v           (8 KB)
    __shared__ float sTmp[64][96];    // stage-1 result         (24 KB)
    __shared__ float sBuSum[64];
    __shared__ float sBvSum[64];
    __shared__ float sRed[256];
    __shared__ int   sRow[32];
    __shared__ int   sCol[32];
    __shared__ int   sMu, sMv, sNan;

    const int b    = blockIdx.x;
    const int t    = threadIdx.x;
    const int lane = t & 31;
    const int wave = t >> 5;
    const int lo   = lane & 15;
    const int hi2  = (lane >> 4) * 2;   // A/B fragment K split
    const int hi8  = (lane >> 4) * 8;   // C/D fragment M split

    // ----- ragged sizes mu, mv from mask
    const float* mk = mask + b * (MU_MAX * MU_MAX);
    if (t == 0) sNan = 0;
    if (t < 32) {
        int any = 0;
        for (int j = 0; j < 32; ++j) any |= (mk[t * 32 + j] > 0.0f) ? 1 : 0;
        sRow[t] = any;
    } else if (t < 64) {
        int c = t - 32, any = 0;
        for (int i = 0; i < 32; ++i) any |= (mk[i * 32 + c] > 0.0f) ? 1 : 0;
        sCol[c] = any;
    }
    __syncthreads();
    if (t == 0) {
        int cu = 0, cv = 0;
        for (int i = 0; i < 32; ++i) { cu += sRow[i]; cv += sCol[i]; }
        sMu = cu < (P_DEG + 1) ? (P_DEG + 1) : cu;
        sMv = cv < (P_DEG + 1) ? (P_DEG + 1) : cv;
    }
    __syncthreads();

    // ----- Cox-de Boor basis: threads 0..63 -> Bu rows, 64..127 -> Bv rows
    if (t < 128) {
        const int   isV = t >> 6;
        const int   idx = t & 63;
        const int   m   = isV ? sMv : sMu;           // valid ctrl count
        const float u   = SURF_EPS + (1.0f - 2.0f * SURF_EPS) * (float)idx / 63.0f;
        // clamped open-uniform knots, padded to 36; L = m + deg + 1
        float kv[36];
        const float Lm1 = (float)(m + P_DEG);        // L - 1
        for (int k = 0; k < 36; ++k) {
            float val = (float)k / Lm1;
            if (k <= P_DEG)  val = 0.0f;
            else if (k >= m) val = 1.0f;             // k >= L-1-deg
            kv[k] = val;
        }
        float N[35];
        for (int k = 0; k < 35; ++k)
            N[k] = (u >= kv[k] && u < kv[k + 1]) ? 1.0f : 0.0f;
        for (int d = 1; d <= P_DEG; ++d) {
            for (int k = 0; k < 35 - d; ++k) {
                float den1 = kv[k + d]     - kv[k];
                float den2 = kv[k + d + 1] - kv[k + 1];
                float s1 = (den1 > 0.0f) ? (u - kv[k]) / den1 : 0.0f;
                float s2 = (den2 > 0.0f) ? (kv[k + d + 1] - u) / den2 : 0.0f;
                N[k] = s1 * N[k] + s2 * N[k + 1];    // in-place forward is safe
            }
        }
        float sum = 0.0f;
        if (isV) {
            for (int i = 0; i < 32; ++i) { sBv[idx][i] = N[i]; sum += N[i]; }
            sBvSum[idx] = sum;
        } else {
            for (int i = 0; i < 32; ++i) { sBu[idx][i] = N[i]; sum += N[i]; }
            sBuSum[idx] = sum;
        }
    }
    __syncthreads();

    // ----- stage 1: tmp[64x96] = Bu[64x32] x ctrlB[32x96]
    // ctrl[b] is contiguous (32,96) row-major: ctrlB[i][j*3+d]
    const float* cb = ctrl + b * (MU_MAX * MU_MAX * 3);
    for (int q = 0; q < 3; ++q) {                    // 24 tiles / 8 waves
        const int T  = wave * 3 + q;
        const int mT = T / 6;                        // u tile (4)
        const int nT = T % 6;                        // col tile (6)
        v8f acc = {};
        for (int k0 = 0; k0 < 32; k0 += 4) {
            v2f a, bm;
            // A 16x4 fragment: lane L -> M=L%16, K=k0+(L/16)*2+r
            a.x = sBu[mT * 16 + lo][k0 + hi2];
            a.y = sBu[mT * 16 + lo][k0 + hi2 + 1];
            // B 4x16 fragment: reg r -> K=k0+(L/16)*2+r, N=nT*16+L%16
            bm.x = cb[(k0 + hi2)     * 96 + nT * 16 + lo];
            bm.y = cb[(k0 + hi2 + 1) * 96 + nT * 16 + lo];
            acc = __builtin_amdgcn_wmma_f32_16x16x4_f32(
                    false, a, false, bm, (short)0, acc, false, false);
        }
        // D layout: reg r, lanes0-15 -> M=r, lanes16-31 -> M=r+8
        for (int r = 0; r < 8; ++r)
            sTmp[mT * 16 + hi8 + r][nT * 16 + lo] = acc[r];
    }
    __syncthreads();

    // ----- stage 2: per channel d, surf[64x64] = tmpD[64x32] x BvT[32x64]
    float localSum = 0.0f;
    int   localNan = 0;
    const float* xb = xyz + b * (OUT_N * OUT_N * 3);
    for (int q = 0; q < 6; ++q) {                    // 48 tiles / 8 waves
        const int T   = wave * 6 + q;
        const int d   = T >> 4;                      // channel
        const int rem = T & 15;
        const int mT  = rem >> 2;                    // u tile
        const int nT  = rem & 3;                     // v tile
        v8f acc = {};
        for (int k0 = 0; k0 < 32; k0 += 4) {
            v2f a, bm;
            const int ka = k0 + hi2;
            a.x = sTmp[mT * 16 + lo][ka * 3 + d];
            a.y = sTmp[mT * 16 + lo][(ka + 1) * 3 + d];
            bm.x = sBv[nT * 16 + lo][ka];            // B[j][v] = Bv[v][j]
            bm.y = sBv[nT * 16 + lo][ka + 1];
            acc = __builtin_amdgcn_wmma_f32_16x16x4_f32(
                    false, a, false, bm, (short)0, acc, false, false);
        }
        for (int r = 0; r < 8; ++r) {
            const int u = mT * 16 + hi8 + r;
            const int v = nT * 16 + lo;
            const float s = acc[r] / (sBuSum[u] * sBvSum[v]);
            if (!__builtin_isfinite(s)) localNan = 1;
            const float diff = s - xb[(u * OUT_N + v) * 3 + d];
            localSum += diff * diff;
        }
    }
    if (localNan) sNan = 1;                          // benign race
    sRed[t] = localSum;
    __syncthreads();
    for (int s = 128; s > 0; s >>= 1) {
        if (t < s) sRed[t] += sRed[t + s];
        __syncthreads();
    }
    if (t == 0) {
        atomicAdd(&ws[2], sRed[0]);
        if (sNan) atomicOr((int*)(ws + 3), 1);
    }
}

// ------------------------------------------------------------- finalize
__global__ void finalize_kernel(const float* __restrict__ ws,
                                float* __restrict__ out) {
    if (threadIdx.x == 0 && blockIdx.x == 0) {
        const float ctrlNum = ws[0];
        const float maskSum = ws[1];
        const float surfSum = ws[2];
        const int   nan     = ((const int*)ws)[3];
        float denomC = maskSum * 3.0f;
        if (denomC < 1.0f) denomC = 1.0f;
        const float loss_ctrl = ctrlNum / denomC;
        const float loss_surf = surfSum / (float)(B_SZ * OUT_N * OUT_N * 3);
        out[0] = loss_ctrl + loss_surf;              // total (pre nan-guard)
        out[1] = loss_ctrl;
        out[2] = nan ? 1.0e6f : loss_surf;
    }
}

// ------------------------------------------------------------- launcher
extern "C" void kernel_launch(void* const* d_in, const int* in_sizes, int n_in,
                              void* d_out, int out_size, void* d_ws, size_t ws_size,
                              hipStream_t stream) {
    (void)in_sizes; (void)n_in; (void)out_size; (void)ws_size;
    const float* pred = (const float*)d_in[0];   // (B,32,32,3)
    const float* gt   = (const float*)d_in[1];   // (B,32,32,3)
    const float* mask = (const float*)d_in[2];   // (B,32,32)
    const float* xyz  = (const float*)d_in[3];   // (B,64,64,3)
    float* ws  = (float*)d_ws;
    float* out = (float*)d_out;

    init_ws_kernel<<<1, 32, 0, stream>>>(ws);
    ctrl_loss_kernel<<<256, 256, 0, stream>>>(pred, gt, mask, ws);
    surf_loss_kernel<<<B_SZ, 256, 0, stream>>>(pred, mask, xyz, ws);
    finalize_kernel<<<1, 32, 0, stream>>>(ws, out);
}